// Attention_4088808866132
// MI455X (gfx1250) — compile-verified
//
#include <hip/hip_runtime.h>
#include <hip/hip_bf16.h>

#define BATCH 16
#define SEQ   4096
#define EMB   1024
#define HEADS 16
#define KSEL  2048   // N * (1 - PRUNE_RATIO)
#define HD    64
#define SCALE 0.125f          // 1/sqrt(64)
#define NOISE_F 0.05f         // NOISE_SCALE * 0.5

typedef __attribute__((ext_vector_type(2))) float v2f;
typedef __attribute__((ext_vector_type(8))) float v8f;

// ---------------------------------------------------------------------------
// Stage 1: qh[b, j] = q_global[b,:] . Wq[j,:] + bq[j]      (B*E outputs)
// ---------------------------------------------------------------------------
__global__ void qh_kernel(const float* __restrict__ qg, const float* __restrict__ in_w,
                          const float* __restrict__ in_b, float* __restrict__ qh) {
    int i = blockIdx.x * 256 + threadIdx.x;      // B*E
    int b = i >> 10, j = i & 1023;
    const float* q = qg + (size_t)b * EMB;
    const float* w = in_w + (size_t)j * EMB;     // Wq row j
    float acc = in_b[j];
    #pragma unroll 4
    for (int e = 0; e < EMB; e += 4) {
        float4 qv = *(const float4*)(q + e);
        float4 wv = *(const float4*)(w + e);
        acc += qv.x * wv.x + qv.y * wv.y + qv.z * wv.z + qv.w * wv.w;
    }
    qh[i] = acc;
}

// ---------------------------------------------------------------------------
// Stage 2: u[b,h,e] = sum_d qh[b, h*64+d] * Wk[h*64+d, e]   (B*H*E outputs)
// ---------------------------------------------------------------------------
__global__ void u_kernel(const float* __restrict__ qh, const float* __restrict__ in_w,
                         float* __restrict__ u) {
    int i = blockIdx.x * 256 + threadIdx.x;      // B*H*E
    int e = i & 1023;
    int bh = i >> 10, b = bh >> 4, h = bh & 15;
    const float* q  = qh + (size_t)b * EMB + h * HD;
    const float* wk = in_w + ((size_t)(EMB + h * HD)) * EMB + e;  // Wk rows, stride EMB
    float acc = 0.f;
    #pragma unroll 8
    for (int d = 0; d < HD; ++d) acc += q[d] * wk[(size_t)d * EMB];
    u[i] = acc;
}

// sconst[b,h] = sum_d qh[b,h*64+d] * bk[h*64+d]
__global__ void sconst_kernel(const float* __restrict__ qh, const float* __restrict__ in_b,
                              float* __restrict__ sconst) {
    int bh = threadIdx.x;                        // 256 = B*H
    int b = bh >> 4, h = bh & 15;
    const float* q  = qh + (size_t)b * EMB + h * HD;
    const float* bk = in_b + EMB + h * HD;
    float acc = 0.f;
    #pragma unroll 8
    for (int d = 0; d < HD; ++d) acc += q[d] * bk[d];
    sconst[bh] = acc;
}

// ---------------------------------------------------------------------------
// WMMA f32 16x16x4 scores:  tile = 16 seq rows x 16 heads, K-loop over EMB.
// A frag (16x4 f32): lane l -> row l&15, cols k + 2*(l>=16) + {0,1}
// B frag (4x16 f32): lane l -> col (head) l&15, rows k + 2*(l>=16) + {0,1}
// D: vgpr r -> row r + 8*(l>=16), col l&15
// ---------------------------------------------------------------------------
__global__ void scores1_wmma(const float* __restrict__ v_seq, const float* __restrict__ u,
                             const float* __restrict__ sconst, float* __restrict__ s1) {
    int b    = blockIdx.y;
    int wave = threadIdx.x >> 5;
    int lane = threadIdx.x & 31;
    int n0   = blockIdx.x * 64 + wave * 16;
    int m    = lane & 15;
    int khalf = (lane >> 4) << 1;
    const float* arow = v_seq + ((size_t)(b * SEQ + n0 + m)) * EMB + khalf;
    const float* brow = u + ((size_t)(b * HEADS + m)) * EMB + khalf;
    v8f acc = {};
    #pragma unroll 8
    for (int k = 0; k < EMB; k += 4) {
        v2f a  = *(const v2f*)(arow + k);
        v2f bb = *(const v2f*)(brow + k);
        acc = __builtin_amdgcn_wmma_f32_16x16x4_f32(false, a, false, bb,
                                                    (short)0, acc, false, false);
    }
    float sc = sconst[b * HEADS + m];
    int rbase = (lane >> 4) * 8;
    #pragma unroll
    for (int r = 0; r < 8; ++r) {
        int n = n0 + rbase + r;
        s1[((size_t)(b * SEQ + n)) * HEADS + m] = (acc[r] + sc) * SCALE;
    }
}

__global__ void scores2_wmma(const float* __restrict__ v_seq, const float* __restrict__ u,
                             const float* __restrict__ sconst, const int* __restrict__ idx,
                             float* __restrict__ s2) {
    int b    = blockIdx.y;
    int wave = threadIdx.x >> 5;
    int lane = threadIdx.x & 31;
    int n0   = blockIdx.x * 64 + wave * 16;
    int m    = lane & 15;
    int khalf = (lane >> 4) << 1;
    int row  = idx[b * KSEL + n0 + m];
    const float* arow = v_seq + ((size_t)(b * SEQ + row)) * EMB + khalf;
    const float* brow = u + ((size_t)(b * HEADS + m)) * EMB + khalf;
    v8f acc = {};
    #pragma unroll 8
    for (int k = 0; k < EMB; k += 4) {
        v2f a  = *(const v2f*)(arow + k);
        v2f bb = *(const v2f*)(brow + k);
        acc = __builtin_amdgcn_wmma_f32_16x16x4_f32(false, a, false, bb,
                                                    (short)0, acc, false, false);
    }
    float sc = sconst[b * HEADS + m];
    int rbase = (lane >> 4) * 8;
    #pragma unroll
    for (int r = 0; r < 8; ++r) {
        int n = n0 + rbase + r;
        s2[((size_t)(b * KSEL + n)) * HEADS + m] = (acc[r] + sc) * SCALE;
    }
}

// ---------------------------------------------------------------------------
// Softmax stats over seq for each (b,h): max + sum(exp)
// ---------------------------------------------------------------------------
__global__ void softmax_stats1(const float* __restrict__ s1, float* __restrict__ m1,
                               float* __restrict__ z1) {
    int bh = blockIdx.x;
    int b = bh >> 4, h = bh & 15;
    __shared__ float red[256];
    const float* base = s1 + (size_t)b * SEQ * HEADS + h;
    float mx = -1e30f;
    for (int n = threadIdx.x; n < SEQ; n += 256)
        mx = fmaxf(mx, base[(size_t)n * HEADS]);
    red[threadIdx.x] = mx; __syncthreads();
    for (int s = 128; s; s >>= 1) {
        if (threadIdx.x < s) red[threadIdx.x] = fmaxf(red[threadIdx.x], red[threadIdx.x + s]);
        __syncthreads();
    }
    mx = red[0]; __syncthreads();
    float sum = 0.f;
    for (int n = threadIdx.x; n < SEQ; n += 256)
        sum += __expf(base[(size_t)n * HEADS] - mx);
    red[threadIdx.x] = sum; __syncthreads();
    for (int s = 128; s; s >>= 1) {
        if (threadIdx.x < s) red[threadIdx.x] += red[threadIdx.x + s];
        __syncthreads();
    }
    if (threadIdx.x == 0) { m1[bh] = mx; z1[bh] = red[0]; }
}

// key[b,n] = mean_h softmax1 + noise*0.05
__global__ void make_keys(const float* __restrict__ s1, const float* __restrict__ m1,
                          const float* __restrict__ z1, const float* __restrict__ noise,
                          float* __restrict__ key) {
    int i = blockIdx.x * 256 + threadIdx.x;      // B*SEQ
    int b = i / SEQ;
    const float* row = s1 + (size_t)i * HEADS;
    float acc = 0.f;
    #pragma unroll
    for (int h = 0; h < HEADS; ++h)
        acc += __expf(row[h] - m1[b * HEADS + h]) / z1[b * HEADS + h];
    key[i] = acc * (1.0f / HEADS) + noise[i] * NOISE_F;
}

// ---------------------------------------------------------------------------
// Exact top-K selection per batch: 4-pass radix select on monotone uint keys.
// Order of selected indices does not matter (second MHA is permutation-inv).
// ---------------------------------------------------------------------------
__device__ __forceinline__ unsigned orderKey(float f) {
    unsigned u = __float_as_uint(f);
    return (u & 0x80000000u) ? ~u : (u | 0x80000000u);
}

__global__ void topk_select(const float* __restrict__ key, int* __restrict__ idx) {
    int b = blockIdx.x;
    __shared__ unsigned hist[256];
    __shared__ unsigned sh_prefix;
    __shared__ int sh_rem, sh_cntHigh, sh_cntTie;
    const float* kb = key + (size_t)b * SEQ;
    if (threadIdx.x == 0) { sh_prefix = 0u; sh_rem = KSEL; sh_cntHigh = 0; sh_cntTie = 0; }
    for (int pass = 3; pass >= 0; --pass) {
        hist[threadIdx.x] = 0u; __syncthreads();
        unsigned prefmask = (pass == 3) ? 0u : (0xFFFFFFFFu << ((pass + 1) * 8));
        unsigned pref = sh_prefix;
        for (int n = threadIdx.x; n < SEQ; n += 256) {
            unsigned u = orderKey(kb[n]);
            if ((u & prefmask) == (pref & prefmask))
                atomicAdd(&hist[(u >> (pass * 8)) & 255], 1u);
        }
        __syncthreads();
        if (threadIdx.x == 0) {
            int rem = sh_rem; unsigned cum = 0;
            for (int bin = 255; bin >= 0; --bin) {
                unsigned c = hist[bin];
                if (cum + c >= (unsigned)rem) {
                    sh_prefix = pref | ((unsigned)bin << (pass * 8));
                    sh_rem = rem - (int)cum;
                    break;
                }
                cum += c;
            }
        }
        __syncthreads();
    }
    unsigned T = sh_prefix;
    int rem = sh_rem;                    // #ties to take; count(>T) == KSEL-rem
    int* ob = idx + (size_t)b * KSEL;
    for (int n = threadIdx.x; n < SEQ; n += 256) {
        unsigned u = orderKey(kb[n]);
        if (u > T) {
            int p = atomicAdd(&sh_cntHigh, 1);
            ob[p] = n;
        } else if (u == T) {
            int p = atomicAdd(&sh_cntTie, 1);
            if (p < rem) ob[KSEL - rem + p] = n;
        }
    }
}

// ---------------------------------------------------------------------------
// Softmax over selected rows (in place): s2 -> p2
// ---------------------------------------------------------------------------
__global__ void softmax2(float* __restrict__ s2) {
    int bh = blockIdx.x;
    int b = bh >> 4, h = bh & 15;
    __shared__ float red[256];
    float* base = s2 + (size_t)b * KSEL * HEADS + h;
    float mx = -1e30f;
    for (int k = threadIdx.x; k < KSEL; k += 256)
        mx = fmaxf(mx, base[(size_t)k * HEADS]);
    red[threadIdx.x] = mx; __syncthreads();
    for (int s = 128; s; s >>= 1) {
        if (threadIdx.x < s) red[threadIdx.x] = fmaxf(red[threadIdx.x], red[threadIdx.x + s]);
        __syncthreads();
    }
    mx = red[0]; __syncthreads();
    float sum = 0.f;
    for (int k = threadIdx.x; k < KSEL; k += 256)
        sum += __expf(base[(size_t)k * HEADS] - mx);
    red[threadIdx.x] = sum; __syncthreads();
    for (int s = 128; s; s >>= 1) {
        if (threadIdx.x < s) red[threadIdx.x] += red[threadIdx.x + s];
        __syncthreads();
    }
    float inv = 1.0f / red[0];
    for (int k = threadIdx.x; k < KSEL; k += 256)
        base[(size_t)k * HEADS] = __expf(base[(size_t)k * HEADS] - mx) * inv;
}

// ---------------------------------------------------------------------------
// m[b,h,e] = sum_k p2[b,h,k] * v_seq[b, idx[k], e]  (split over 4 K-partitions)
// ---------------------------------------------------------------------------
__global__ void weighted_sum(const float* __restrict__ v_seq, const float* __restrict__ p2,
                             const int* __restrict__ idx, float* __restrict__ partial) {
    int b = blockIdx.y;
    int part = blockIdx.z;
    int e = blockIdx.x * 256 + threadIdx.x;
    __shared__ float shp[64 * HEADS];
    __shared__ int   shi[64];
    float acc[HEADS];
    #pragma unroll
    for (int h = 0; h < HEADS; ++h) acc[h] = 0.f;
    int kbeg = part * (KSEL / 4), kend = kbeg + (KSEL / 4);
    for (int k0 = kbeg; k0 < kend; k0 += 64) {
        __syncthreads();
        #pragma unroll
        for (int t = 0; t < 4; ++t)
            shp[threadIdx.x + t * 256] =
                p2[((size_t)(b * KSEL + k0)) * HEADS + threadIdx.x + t * 256];
        if (threadIdx.x < 64) shi[threadIdx.x] = idx[b * KSEL + k0 + threadIdx.x];
        __syncthreads();
        for (int j = 0; j < 64; ++j) {
            float v = v_seq[((size_t)(b * SEQ + shi[j])) * EMB + e];
            #pragma unroll
            for (int h = 0; h < HEADS; ++h) acc[h] += shp[j * HEADS + h] * v;
        }
    }
    #pragma unroll
    for (int h = 0; h < HEADS; ++h)
        partial[((size_t)part * BATCH * HEADS + b * HEADS + h) * EMB + e] = acc[h];
}

__global__ void reduce_parts(const float* __restrict__ partial, float* __restrict__ m) {
    int i = blockIdx.x * 256 + threadIdx.x;      // B*H*E
    const size_t stride = (size_t)BATCH * HEADS * EMB;
    m[i] = partial[i] + partial[i + stride] + partial[i + 2 * stride] + partial[i + 3 * stride];
}

// ctx[b, h*64+d] = m[b,h,:] . Wv[h*64+d,:] + bv[h*64+d]
__global__ void ctx_kernel(const float* __restrict__ m, const float* __restrict__ in_w,
                           const float* __restrict__ in_b, float* __restrict__ ctx) {
    int i = blockIdx.x * 256 + threadIdx.x;      // B*E
    int b = i >> 10, r = i & 1023;
    int h = r >> 6;
    const float* mrow = m + ((size_t)(b * HEADS + h)) * EMB;
    const float* wrow = in_w + ((size_t)(2 * EMB + r)) * EMB;
    float acc = in_b[2 * EMB + r];
    #pragma unroll 4
    for (int e = 0; e < EMB; e += 4) {
        float4 mv = *(const float4*)(mrow + e);
        float4 wv = *(const float4*)(wrow + e);
        acc += mv.x * wv.x + mv.y * wv.y + mv.z * wv.z + mv.w * wv.w;
    }
    ctx[i] = acc;
}

// out[b, e] = ctx[b,:] . out_w[e,:] + out_b[e];  out[b, E+e] = q_global[b,e]
__global__ void out_kernel(const float* __restrict__ ctx, const float* __restrict__ out_w,
                           const float* __restrict__ out_b, const float* __restrict__ qg,
                           float* __restrict__ out) {
    int i = blockIdx.x * 256 + threadIdx.x;      // B*E
    int b = i >> 10, e = i & 1023;
    const float* c = ctx + (size_t)b * EMB;
    const float* w = out_w + (size_t)e * EMB;
    float acc = out_b[e];
    #pragma unroll 4
    for (int k = 0; k < EMB; k += 4) {
        float4 cv = *(const float4*)(c + k);
        float4 wv = *(const float4*)(w + k);
        acc += cv.x * wv.x + cv.y * wv.y + cv.z * wv.z + cv.w * wv.w;
    }
    out[(size_t)b * 2 * EMB + e] = acc;
    out[(size_t)b * 2 * EMB + EMB + e] = qg[i];
}

// ---------------------------------------------------------------------------
extern "C" void kernel_launch(void* const* d_in, const int* in_sizes, int n_in,
                              void* d_out, int out_size, void* d_ws, size_t ws_size,
                              hipStream_t stream) {
    const float* v_seq    = (const float*)d_in[0];
    const float* q_global = (const float*)d_in[3];
    const float* noise    = (const float*)d_in[4];
    const float* in_w     = (const float*)d_in[5];
    const float* in_b     = (const float*)d_in[6];
    const float* out_w    = (const float*)d_in[7];
    const float* out_b    = (const float*)d_in[8];
    float* out = (float*)d_out;

    float* ws = (float*)d_ws;
    size_t off = 0;
    float* qh     = ws + off; off += (size_t)BATCH * EMB;           // 16384
    float* u      = ws + off; off += (size_t)BATCH * HEADS * EMB;   // 262144
    float* sconst = ws + off; off += BATCH * HEADS;                 // 256
    float* s1     = ws + off; off += (size_t)BATCH * SEQ * HEADS;   // 1048576 (reused as partial)
    float* m1     = ws + off; off += BATCH * HEADS;
    float* z1     = ws + off; off += BATCH * HEADS;
    float* key    = ws + off; off += (size_t)BATCH * SEQ;           // 65536
    int*   idx    = (int*)(ws + off); off += (size_t)BATCH * KSEL;  // 32768
    float* p2     = ws + off; off += (size_t)BATCH * KSEL * HEADS;  // 524288
    float* mbuf   = ws + off; off += (size_t)BATCH * HEADS * EMB;   // 262144
    float* ctx    = ws + off; off += (size_t)BATCH * EMB;           // 16384
    float* partial = s1;  // s1 no longer needed after make_keys; 4*262144 fits in s1

    qh_kernel<<<BATCH * EMB / 256, 256, 0, stream>>>(q_global, in_w, in_b, qh);
    u_kernel<<<BATCH * HEADS * EMB / 256, 256, 0, stream>>>(qh, in_w, u);
    sconst_kernel<<<1, 256, 0, stream>>>(qh, in_b, sconst);
    scores1_wmma<<<dim3(SEQ / 64, BATCH), 128, 0, stream>>>(v_seq, u, sconst, s1);
    softmax_stats1<<<BATCH * HEADS, 256, 0, stream>>>(s1, m1, z1);
    make_keys<<<BATCH * SEQ / 256, 256, 0, stream>>>(s1, m1, z1, noise, key);
    topk_select<<<BATCH, 256, 0, stream>>>(key, idx);
    scores2_wmma<<<dim3(KSEL / 64, BATCH), 128, 0, stream>>>(v_seq, u, sconst, idx, p2);
    softmax2<<<BATCH * HEADS, 256, 0, stream>>>(p2);
    weighted_sum<<<dim3(EMB / 256, BATCH, 4), 256, 0, stream>>>(v_seq, p2, idx, partial);
    reduce_parts<<<BATCH * HEADS * EMB / 256, 256, 0, stream>>>(partial, mbuf);
    ctx_kernel<<<BATCH * EMB / 256, 256, 0, stream>>>(mbuf, in_w, in_b, ctx);
    out_kernel<<<BATCH * EMB / 256, 256, 0, stream>>>(ctx, out_w, out_b, q_global, out);
}